// RandomGaussianBlur_45475113730338
// MI455X (gfx1250) — compile-verified
//
#include <hip/hip_runtime.h>
#include <cstdint>

// RandomGaussianBlur: B=32, H=512, W=512, C=3, fp32 NHWC, per-image sigma.
// Memory-bound (AI ~1.5 flop/byte -> ~8.6us floor at 23.3 TB/s for 201 MB).
// CDNA5 path: Tensor Data Mover DMAs (tensor_load_to_lds /
// tensor_store_from_lds, TENSORcnt) stage contiguous row strips in LDS;
// separable 3-tap blur runs in LDS with fp32 VALU for exact parity.

#define BLK_T    256
#define TILE_H   8
#define ROWF     1536                     // W*C floats per image row
#define ROWB     (ROWF * 4)               // bytes per image row (6144)
#define HALO_F   ((TILE_H + 2) * ROWF)    // 15360 floats staged per block
#define OUT_F    (TILE_H * ROWF)          // 12288 floats produced per block
#define N4_LOAD  (HALO_F / 4)             // 3840 float4 loads (fallback path)
#define N4_OUT   (OUT_F / 4)              // 3072 float4 outputs
#define IMGF     (512 * ROWF)             // floats per image
#define ROW4     (ROWF / 4)               // 384 float4 per row

typedef unsigned int v4u __attribute__((ext_vector_type(4)));
typedef int          v4i __attribute__((ext_vector_type(4)));
typedef int          v8i __attribute__((ext_vector_type(8)));

#if defined(__has_builtin)
#if __has_builtin(__builtin_amdgcn_tensor_load_to_lds) && \
    __has_builtin(__builtin_amdgcn_tensor_store_from_lds)
#define USE_TDM 1
#endif
#endif

#ifdef USE_TDM
// ---- Tensor DMA descriptor helpers (1-D tile, 4-byte elements) ----
// Group 0: [31:0]=count(=1); [63:32]=lds byte addr; [120:64]=global addr;
//          [127:126]=type(=2 "image").
__device__ __forceinline__ v4u tdm_g0(unsigned lds_addr, uint64_t gaddr) {
  v4u g0 = { 1u,
             lds_addr,
             (unsigned)(gaddr & 0xFFFFFFFFu),
             (unsigned)(gaddr >> 32) | 0x80000000u };
  return g0;
}
// Group 1: data_size=2 (4B); tensor_dim0=td0 (OOB reads past it return zero,
// which implements the 'SAME' zero padding at the image bottom edge);
// tensor_dim1=1; tile_dim0=tile0 (1-D tile: tile_dim1=tile_dim2=0);
// tensor_dim0_stride=td0; no multicast / iterate / pad / atomic-barrier.
__device__ __forceinline__ v8i tdm_g1(unsigned td0, unsigned tile0) {
  v8i g1 = { (int)0x00020000u,
             (int)((td0 & 0xFFFFu) << 16),
             (int)((td0 >> 16) | (1u << 16)),
             (int)(tile0 << 16),
             0,
             (int)td0,
             0, 0 };
  return g1;
}

__device__ __forceinline__ void tdm_load(v4u g0, v8i g1) {
  v4i z4 = {0, 0, 0, 0};
#if __clang_major__ >= 23
  v8i z8 = {0, 0, 0, 0, 0, 0, 0, 0};
  __builtin_amdgcn_tensor_load_to_lds(g0, g1, z4, z4, z8, 0);
#else
  __builtin_amdgcn_tensor_load_to_lds(g0, g1, z4, z4, 0);
#endif
}
__device__ __forceinline__ void tdm_store(v4u g0, v8i g1) {
  v4i z4 = {0, 0, 0, 0};
#if __clang_major__ >= 23
  v8i z8 = {0, 0, 0, 0, 0, 0, 0, 0};
  __builtin_amdgcn_tensor_store_from_lds(g0, g1, z4, z4, z8, 0);
#else
  __builtin_amdgcn_tensor_store_from_lds(g0, g1, z4, z4, 0);
#endif
}
#endif // USE_TDM

__global__ __launch_bounds__(BLK_T)
void gblur_separable_kernel(const float* __restrict__ x,
                            const float* __restrict__ sigmas,
                            float* __restrict__ out) {
  extern __shared__ float smem[];
  float* buf1 = smem;            // [10][1536] input halo strip (reused as out stage)
  float* buf2 = smem + HALO_F;   // [ 8][1536] vertically blurred (+pad)

  const int tid  = threadIdx.x;
  const int img  = blockIdx.x >> 6;          // 64 strips per image
  const int row0 = (blockIdx.x & 63) * TILE_H;

  const float* imgp = x + (size_t)img * IMGF;

  // Separable weights matching the reference: std = sigma + 1e-5,
  // e = exp(-1/(2 std^2)); 1D kernel [e,1,e]/(1+2e).
  const float stdv = sigmas[img] + 1e-5f;
  const float ev   = expf(-1.0f / (2.0f * stdv * stdv));
  const float w0   = 1.0f / (1.0f + 2.0f * ev);
  const float w1   = ev * w0;

  // Low 32 bits of a generic LDS pointer == byte offset from wave LDS base.
  const unsigned lds0 = (unsigned)(uintptr_t)smem;

#ifdef USE_TDM
  // Scalar (wave-uniform) predicate: TDM ignores EXEC, so gate by wave id
  // via readfirstlane to get a truly scalar branch (exactly one DMA issue).
  const int wfirst = __builtin_amdgcn_readfirstlane((int)threadIdx.x);

  // ---- Phase 1: one TDM DMA of the 10-row halo strip into LDS ----
  // Strip is contiguous in NHWC. Bottom-edge zero padding falls out of the
  // D# tensor_dim0 OOB rule; the top halo row of strip 0 is zeroed manually.
  if (wfirst == 0) {                           // wave 0 only
    const unsigned gstart = (row0 == 0) ? 0u : (unsigned)(row0 - 1);
    const unsigned dst    = (row0 == 0) ? (unsigned)ROWB : 0u;
    const unsigned tile0  = (row0 == 0) ? (unsigned)(HALO_F - ROWF)
                                        : (unsigned)HALO_F;
    const unsigned td0    = (512u - gstart) * (unsigned)ROWF;
    const uint64_t ga     = (uint64_t)(uintptr_t)imgp + (uint64_t)gstart * ROWB;
    tdm_load(tdm_g0(lds0 + dst, ga), tdm_g1(td0, tile0));
    __builtin_amdgcn_s_wait_tensorcnt(0);
  }
  if (row0 == 0) {
    for (int q = tid; q < ROW4; q += BLK_T)    // zero the top halo row
      *(float4*)&buf1[q * 4] = make_float4(0.f, 0.f, 0.f, 0.f);
  }
  __syncthreads();
#else
  // ---- Fallback: per-lane async global->LDS copies (ASYNCcnt) ----
  for (int q = tid; q < N4_LOAD; q += BLK_T) {
    const int r = q / ROW4;
    const int g = row0 - 1 + r;
    if ((unsigned)g < 512u) {
      const unsigned loff = lds0 + (unsigned)q * 16u;
      const unsigned voff = (unsigned)g * (unsigned)ROWB
                          + (unsigned)(q - r * ROW4) * 16u;
      asm volatile("global_load_async_to_lds_b128 %0, %1, %2"
                   :: "v"(loff), "v"(voff), "s"(imgp) : "memory");
    } else {
      *(float4*)&buf1[q * 4] = make_float4(0.f, 0.f, 0.f, 0.f);
    }
  }
  asm volatile("s_wait_asynccnt 0x0" ::: "memory");
  __syncthreads();
#endif

  // ---- Phase 2: vertical 3-tap (float4, aligned, bank-conflict free) ----
  for (int q = tid; q < N4_OUT; q += BLK_T) {
    const int f = q * 4;                       // buf2 index; buf1 center = f+ROWF
    const float4 a = *(const float4*)&buf1[f];
    const float4 b = *(const float4*)&buf1[f + ROWF];
    const float4 c = *(const float4*)&buf1[f + 2 * ROWF];
    float4 v;
    v.x = w0 * b.x + w1 * (a.x + c.x);
    v.y = w0 * b.y + w1 * (a.y + c.y);
    v.z = w0 * b.z + w1 * (a.z + c.z);
    v.w = w0 * b.w + w1 * (a.w + c.w);
    *(float4*)&buf2[f] = v;
  }
  __syncthreads();

  // ---- Phase 3: horizontal 3-tap (+/-3 floats = +/-1 pixel) ----
  for (int q = tid; q < N4_OUT; q += BLK_T) {
    const int  f   = q * 4;
    const int  col = f % ROWF;                 // multiple of 4
    const bool lz  = (col == 0);               // pixel x == 0 in lanes 0..2
    const bool rz  = (col == ROWF - 4);        // pixel x == 511 in lanes 1..3

    const float4 c  = *(const float4*)&buf2[f];
    const float4 A  = *(const float4*)(&buf2[f] - 4);  // f-4..f-1 (aligned)
    const float4 Bv = *(const float4*)(&buf2[f] + 4);  // f+4..f+7 (padded)

    const float L0 = lz ? 0.f : A.y;   // buf2[f-3]
    const float L1 = lz ? 0.f : A.z;   // buf2[f-2]
    const float L2 = lz ? 0.f : A.w;   // buf2[f-1]
    const float L3 = c.x;              // buf2[f]   (always in-row)
    const float R0 = c.w;              // buf2[f+3] (always in-row)
    const float R1 = rz ? 0.f : Bv.x;  // buf2[f+4]
    const float R2 = rz ? 0.f : Bv.y;  // buf2[f+5]
    const float R3 = rz ? 0.f : Bv.z;  // buf2[f+6]

    float4 o;
    o.x = w0 * c.x + w1 * (L0 + R0);
    o.y = w0 * c.y + w1 * (L1 + R1);
    o.z = w0 * c.z + w1 * (L2 + R2);
    o.w = w0 * c.w + w1 * (L3 + R3);

#ifdef USE_TDM
    *(float4*)&buf1[f] = o;            // stage result in LDS (buf1 is dead)
#else
    float* op = out + (size_t)img * IMGF + (size_t)row0 * ROWF;
    *(float4*)&op[f] = o;
#endif
  }

#ifdef USE_TDM
  __syncthreads();
  // ---- Phase 4: one TDM DMA of the 8-row result strip back to HBM ----
  if (wfirst == 0) {
    const uint64_t oa = (uint64_t)(uintptr_t)
        (out + (size_t)img * IMGF + (size_t)row0 * ROWF);
    tdm_store(tdm_g0(lds0, oa), tdm_g1((unsigned)OUT_F, (unsigned)OUT_F));
    __builtin_amdgcn_s_wait_tensorcnt(0);  // complete before LDS teardown
  }
#endif
}

extern "C" void kernel_launch(void* const* d_in, const int* in_sizes, int n_in,
                              void* d_out, int out_size, void* d_ws, size_t ws_size,
                              hipStream_t stream) {
  const float* x      = (const float*)d_in[0];   // [32,512,512,3] fp32
  const float* sigmas = (const float*)d_in[1];   // [32] fp32
  float* out          = (float*)d_out;           // [32,512,512,3] fp32

  const dim3 grid(32 * 64);       // 32 images * 64 row strips
  const dim3 block(BLK_T);
  const size_t shmem = (size_t)(HALO_F + OUT_F + 4) * sizeof(float); // 110,608 B

  gblur_separable_kernel<<<grid, block, shmem, stream>>>(x, sigmas, out);
}